// PatchSimilarityLoss_21526376087966
// MI455X (gfx1250) — compile-verified
//
#include <hip/hip_runtime.h>

// PatchSimilarityLoss for (B,C,H,W) = (32,3,600,600) fp32.
// Memory-bound streaming reduction: ~276.5 MB read -> ~12 us floor @ 23.3 TB/s.
// Strategy: b128 non-temporal loads (2 groups/iter = 12 loads in flight),
// batch index as gridDim.y (no integer division in the hot loop), per-thread
// fp32 accumulation, wave32 shuffle reduction, per-block partials to d_ws,
// single-block f64 finalize.

typedef __attribute__((ext_vector_type(4))) float v4f;

#define HW      360000          // 600*600 (one channel plane)
#define CHW     (3 * HW)        // one image
#define GPB     (HW / 4)        // 90000 float4-groups per image plane
#define NBATCH  32
#define NPIX    ((double)NBATCH * (double)HW)        // sim count
#define NTOT    ((double)NBATCH * 3.0 * (double)HW)  // l1 count

__device__ __forceinline__ float wave_reduce_add(float v) {
    // wave32 on gfx1250
    #pragma unroll
    for (int off = 16; off > 0; off >>= 1)
        v += __shfl_down(v, off, 32);
    return v;
}

__device__ __forceinline__ void accum_group(const float* __restrict__ pb,
                                            const float* __restrict__ tb,
                                            int g, float& simAcc, float& l1Acc) {
    const int base = g << 2;   // element offset, 16B aligned

    // 6 x global_load_b128 th:NT (single-pass 276 MB stream > 192 MB L2).
    v4f p0 = __builtin_nontemporal_load((const v4f*)(pb + base));
    v4f p1 = __builtin_nontemporal_load((const v4f*)(pb + base + HW));
    v4f p2 = __builtin_nontemporal_load((const v4f*)(pb + base + 2 * HW));
    v4f t0 = __builtin_nontemporal_load((const v4f*)(tb + base));
    v4f t1 = __builtin_nontemporal_load((const v4f*)(tb + base + HW));
    v4f t2 = __builtin_nontemporal_load((const v4f*)(tb + base + 2 * HW));

    v4f dot = p0 * t0 + p1 * t1 + p2 * t2;
    v4f pp  = p0 * p0 + p1 * p1 + p2 * p2;
    v4f tt  = t0 * t0 + t1 * t1 + t2 * t2;

    // max(sqrt(x), 1e-8) == sqrt(max(x, 1e-16)); rsq instead of sqrt+div.
    #pragma unroll
    for (int k = 0; k < 4; ++k) {
        float inv = rsqrtf(fmaxf(pp[k], 1e-16f)) *
                    rsqrtf(fmaxf(tt[k], 1e-16f));
        simAcc += dot[k] * inv;
    }

    v4f d0 = p0 - t0, d1 = p1 - t1, d2 = p2 - t2;
    #pragma unroll
    for (int k = 0; k < 4; ++k)
        l1Acc += fabsf(d0[k]) + fabsf(d1[k]) + fabsf(d2[k]);
}

__global__ __launch_bounds__(256)
void psl_partial_kernel(const float* __restrict__ pred,
                        const float* __restrict__ targ,
                        float2* __restrict__ partial) {
    // gridDim.y = batch image index: no division in the hot loop.
    const float* pb = pred + blockIdx.y * CHW;
    const float* tb = targ + blockIdx.y * CHW;

    const int tid    = blockIdx.x * blockDim.x + threadIdx.x;
    const int stride = gridDim.x * blockDim.x;

    float simAcc = 0.0f;
    float l1Acc  = 0.0f;

    // Two groups per iteration -> 12 b128 loads the compiler can clause.
    for (int g = tid; g < GPB; g += 2 * stride) {
        accum_group(pb, tb, g, simAcc, l1Acc);
        const int g2 = g + stride;
        if (g2 < GPB)
            accum_group(pb, tb, g2, simAcc, l1Acc);
    }

    // Intra-wave (wave32) then cross-wave reduction via LDS.
    __shared__ float sSim[8];
    __shared__ float sL1[8];
    const int lane = threadIdx.x & 31;
    const int wid  = threadIdx.x >> 5;

    float wSim = wave_reduce_add(simAcc);
    float wL1  = wave_reduce_add(l1Acc);
    if (lane == 0) { sSim[wid] = wSim; sL1[wid] = wL1; }
    __syncthreads();

    if (wid == 0) {
        const int nw = blockDim.x >> 5;
        float a = (lane < nw) ? sSim[lane] : 0.0f;
        float c = (lane < nw) ? sL1[lane]  : 0.0f;
        a = wave_reduce_add(a);
        c = wave_reduce_add(c);
        if (lane == 0)
            partial[blockIdx.y * gridDim.x + blockIdx.x] = make_float2(a, c);
    }
}

__global__ __launch_bounds__(256)
void psl_final_kernel(const float2* __restrict__ partial, int nblocks,
                      float* __restrict__ out) {
    __shared__ double dSim[256];
    __shared__ double dL1[256];

    double s = 0.0, l = 0.0;
    for (int i = threadIdx.x; i < nblocks; i += blockDim.x) {
        float2 p = partial[i];
        s += (double)p.x;
        l += (double)p.y;
    }
    dSim[threadIdx.x] = s;
    dL1[threadIdx.x]  = l;
    __syncthreads();

    #pragma unroll
    for (int off = 128; off > 0; off >>= 1) {
        if (threadIdx.x < off) {
            dSim[threadIdx.x] += dSim[threadIdx.x + off];
            dL1[threadIdx.x]  += dL1[threadIdx.x + off];
        }
        __syncthreads();
    }

    if (threadIdx.x == 0) {
        double simMean = dSim[0] / NPIX;
        double l1Mean  = dL1[0] / NTOT;
        out[0] = (float)((1.0 - simMean) + 0.1 * l1Mean);
    }
}

extern "C" void kernel_launch(void* const* d_in, const int* in_sizes, int n_in,
                              void* d_out, int out_size, void* d_ws, size_t ws_size,
                              hipStream_t stream) {
    const float* pred = (const float*)d_in[0];
    const float* targ = (const float*)d_in[1];
    float*       out  = (float*)d_out;
    float2*      part = (float2*)d_ws;

    // 64 x-blocks * 32 batch slices = 2048 blocks * 256 threads = 512K threads,
    // ~5.5 groups each; plenty of in-flight b128 loads to saturate HBM while
    // keeping only 16 KB of partials in scratch. Clamp to ws_size for safety.
    int gx = 64;
    while (gx > 1 && (size_t)gx * NBATCH * sizeof(float2) > ws_size)
        gx >>= 1;

    dim3 grid(gx, NBATCH, 1);
    psl_partial_kernel<<<grid, 256, 0, stream>>>(pred, targ, part);
    psl_final_kernel<<<1, 256, 0, stream>>>(part, gx * NBATCH, out);
}